// Superpixel2Pixel_38165079392835
// MI455X (gfx1250) — compile-verified
//
#include <hip/hip_runtime.h>

#ifndef N_SEGM
#define N_SEGM 1024
#endif

typedef int   int32x4  __attribute__((ext_vector_type(4)));
typedef float floatx4  __attribute__((ext_vector_type(4)));

// out[p] = table[idx[p]]  — memory-bound gather.
// CDNA5 path: stage 4 KB table into LDS via async global->LDS DMA (ASYNCcnt),
// then gather from LDS (ds_load_b32) while streaming idx/out with NT b128 ops.
__global__ __launch_bounds__(256) void superpixel2pixel_gather(
    const float* __restrict__ table,   // [N_SEGM] float32
    const int*   __restrict__ idx,     // [H*W] int32 segment ids
    float*       __restrict__ out,     // [H*W] float32
    int n4)                            // (H*W)/4
{
  __shared__ float lut[N_SEGM];

  // ---- Stage the 4 KB gather table into LDS with the CDNA5 async DMA path.
  // 256 threads x 16 B = 4096 B = N_SEGM * 4 bytes.
  {
    unsigned t = threadIdx.x;
    // Low 32 bits of a generic shared-pointer are the LDS byte address
    unsigned lds_addr = (unsigned)(size_t)(&lut[0]) + t * 16u;
    const float* gsrc = table + (size_t)t * 4u;
    asm volatile("global_load_async_to_lds_b128 %0, %1, off"
                 :: "v"(lds_addr), "v"(gsrc)
                 : "memory");
    asm volatile("s_wait_asynccnt 0" ::: "memory");
  }
  __syncthreads();

  const int32x4* __restrict__ idx4 = (const int32x4*)idx;
  floatx4*       __restrict__ out4 = (floatx4*)out;

  int stride = gridDim.x * blockDim.x;
  for (int i = blockIdx.x * blockDim.x + threadIdx.x; i < n4; i += stride) {
    int32x4 v = __builtin_nontemporal_load(&idx4[i]);   // global_load_b128 NT
    floatx4 r;
    r.x = lut[v.x];                                     // ds_load_b32 gathers
    r.y = lut[v.y];
    r.z = lut[v.z];
    r.w = lut[v.w];
    __builtin_nontemporal_store(r, &out4[i]);           // global_store_b128 NT
  }
}

extern "C" void kernel_launch(void* const* d_in, const int* in_sizes, int n_in,
                              void* d_out, int out_size, void* d_ws, size_t ws_size,
                              hipStream_t stream) {
  const float* table = (const float*)d_in[0];   // input: [1024] f32
  const int*   idx   = (const int*)d_in[1];     // segm_img: [4096*4096] int
  float*       out   = (float*)d_out;           // [4096*4096] f32

  int npix = in_sizes[1];
  int n4   = npix >> 2;                         // 4 pixels per thread-iter

  dim3 block(256);
  int needed    = (n4 + 255) / 256;
  int maxBlocks = 4096;                         // cap so LDS staging stays cheap
  dim3 grid(needed < maxBlocks ? needed : maxBlocks);

  hipLaunchKernelGGL(superpixel2pixel_gather, grid, block, 0, stream,
                     table, idx, out, n4);
}